// ButterflyLinear_41463614275882
// MI455X (gfx1250) — compile-verified
//
#include <hip/hip_runtime.h>

typedef __attribute__((ext_vector_type(2))) float v2f;
typedef __attribute__((ext_vector_type(4))) float v4f;
typedef __attribute__((ext_vector_type(8))) float v8f;

#define IN_F   4096
#define OUT_F  1024
#define BATCH  16

// ---------------------------------------------------------------------------
// Kernel 1: stream zeros over the whole 256MB output with NT b128 stores.
// This is the bandwidth-dominant part (~11us of the ~14.4us roofline).
// ---------------------------------------------------------------------------
__global__ __launch_bounds__(256) void butterfly_zero_kernel(v4f* __restrict__ out,
                                                             long n4) {
    long i      = (long)blockIdx.x * blockDim.x + threadIdx.x;
    long stride = (long)gridDim.x * blockDim.x;
    v4f z = {0.f, 0.f, 0.f, 0.f};
    for (; i < n4; i += stride) {
        __builtin_nontemporal_store(z, &out[i]);
    }
}

// ---------------------------------------------------------------------------
// Kernel 2: band compute via V_WMMA_F32_16X16X4_F32.
// One wave per (batch b, 16-row s-block). Band columns of rows s0..s0+15 are
// the contiguous slice W[:, 4*s0 : 4*s0+64], so the band is the banded
// diagonal of a dense 16x64x1024 GEMM -> 4 accumulator tiles, 256 K=4 steps.
//
// A-fragment layout (32-bit A 16x4): lane(ln,half) holds x[s0+ln, k+2*half+{0,1}]
//   -> one contiguous float2 load per lane.
// B-fragment layout (32-bit, 4x16): lane(ln,half) holds W[k+2*half+{0,1}, col0+ln]
// C/D layout: VGPR r on lane(ln,half) = D[M = r + 8*half, N = ln].
// Band element (m, 4*(m&3)+j) lives on lane ln = 4*(m&3)+j, VGPR r = m&7,
// tile t = m>>2  ->  each lane owns exactly 2 band outputs:
//   tile 2*half   element r1      (row m1 = 8*half + r1)
//   tile 2*half+1 element r1+4    (row m2 = m1 + 4),   r1 = ln>>2, j = ln&3.
// ---------------------------------------------------------------------------
__global__ __launch_bounds__(32) void butterfly_wmma_kernel(const float* __restrict__ x,
                                                            const float* __restrict__ W,
                                                            float* __restrict__ out) {
    const int sblk = blockIdx.x;        // 0..63  (s-block)
    const int b    = blockIdx.y;        // 0..15  (batch)
    const int lane = threadIdx.x;       // 0..31
    const int ln   = lane & 15;
    const int half = lane >> 4;
    const int s0   = sblk * 16;
    const int col0 = s0 * 4;            // contiguous band slice start in W

    // Per-lane A base: row s0+ln of x[b], K-offset 2*half (8B aligned).
    const float* xp = x + ((size_t)b * OUT_F + (s0 + ln)) * OUT_F + 2 * half;
    // Per-lane B base: K-row 2*half, column col0+ln.
    const float* wp = W + (size_t)(2 * half) * IN_F + col0 + ln;

    v8f acc0 = {}, acc1 = {}, acc2 = {}, acc3 = {};

    for (int k = 0; k < OUT_F; k += 4) {
        v2f a = *(const v2f*)(xp + k);                 // x[row, k+2h], x[row, k+2h+1]
        const float* wr = wp + (size_t)k * IN_F;
        v2f b0 = { wr[0],  wr[0  + IN_F] };            // N-tile 0: cols col0+ 0..15
        v2f b1 = { wr[16], wr[16 + IN_F] };            // N-tile 1: cols col0+16..31
        v2f b2 = { wr[32], wr[32 + IN_F] };            // N-tile 2: cols col0+32..47
        v2f b3 = { wr[48], wr[48 + IN_F] };            // N-tile 3: cols col0+48..63

        acc0 = __builtin_amdgcn_wmma_f32_16x16x4_f32(false, a, false, b0, (short)0, acc0, false, false);
        acc1 = __builtin_amdgcn_wmma_f32_16x16x4_f32(false, a, false, b1, (short)0, acc1, false, false);
        acc2 = __builtin_amdgcn_wmma_f32_16x16x4_f32(false, a, false, b2, (short)0, acc2, false, false);
        acc3 = __builtin_amdgcn_wmma_f32_16x16x4_f32(false, a, false, b3, (short)0, acc3, false, false);
    }

    // Each lane stores its 2 band outputs.
    const int r1 = ln >> 2;             // 0..3
    const int j  = ln & 3;              // 0..3
    const int m1 = 8 * half + r1;       // row of first owned band element
    const int m2 = m1 + 4;              // row of second owned band element

    v8f cLo = half ? acc2 : acc0;       // tile 2*half
    v8f cHi = half ? acc3 : acc1;       // tile 2*half + 1

    const size_t o1 = ((size_t)b * OUT_F + (s0 + m1)) * IN_F + (size_t)4 * (s0 + m1) + j;
    const size_t o2 = ((size_t)b * OUT_F + (s0 + m2)) * IN_F + (size_t)4 * (s0 + m2) + j;
    out[o1] = cLo[r1];
    out[o2] = cHi[r1 + 4];
}

extern "C" void kernel_launch(void* const* d_in, const int* in_sizes, int n_in,
                              void* d_out, int out_size, void* d_ws, size_t ws_size,
                              hipStream_t stream) {
    const float* x = (const float*)d_in[0];   // (16, 1024, 1024) fp32
    const float* W = (const float*)d_in[1];   // (1024, 4096) fp32
    float* out = (float*)d_out;               // (16, 1024, 4096) fp32

    // Pass 1: zero the full output (out_size = 2^26 floats, 16-byte divisible).
    long n4 = (long)out_size / 4;
    butterfly_zero_kernel<<<4096, 256, 0, stream>>>((v4f*)out, n4);

    // Pass 2: WMMA band compute. grid = (64 s-blocks, 16 batches), 1 wave each.
    dim3 grid(64, 16);
    butterfly_wmma_kernel<<<grid, 32, 0, stream>>>(x, W, out);
}